// RgcnLP_41747082117128
// MI455X (gfx1250) — compile-verified
//
#include <hip/hip_runtime.h>

#define NUM_ENTITIES  100000
#define NUM_RELATIONS 200
#define NUM_BASES     8
#define DIM           128
#define NUM_EDGES     300000
#define NUM_TRIPLES   4096

typedef __attribute__((ext_vector_type(2))) float v2f;
typedef __attribute__((ext_vector_type(8))) float v8f;

// ---------------- workspace layout (bytes) ----------------
#define OFF_W        ((size_t)0)                          // 200*128*128 f32 = 13,107,200
#define OFF_CNT      ((size_t)13107200)                   // 20M bytes packed u8 counts
#define OFF_AGG      ((size_t)33107200)                   // 100000*128 f32 = 51,200,000
#define OFF_META     ((size_t)84307200)
#define META_HIST    0                                    // 200 int
#define META_OFFS    1024                                 // 200 int
#define META_CURS    2048                                 // 200 int
#define META_TOTT    3072                                 // 1 int
#define META_TREL    4096                                 // 5120 int
#define META_TOFF    (4096 + 5120*4)                      // 5120 int
#define META_SORT    (4096 + 2*5120*4)                    // 300000 int
#define MAX_TILES    5120
#define MAX_TILE_BLOCKS 4900                              // ceil(300000/64)+200 = 4888

// W[r] = sum_b coeff[r,b] * bases[b]
__global__ void build_weights(const float* __restrict__ bases,
                              const float* __restrict__ coeff,
                              float* __restrict__ W) {
  int r = blockIdx.x;
  __shared__ float c[NUM_BASES];
  if (threadIdx.x < NUM_BASES) c[threadIdx.x] = coeff[r * NUM_BASES + threadIdx.x];
  __syncthreads();
  for (int i = threadIdx.x; i < DIM * DIM; i += blockDim.x) {
    float acc = 0.f;
#pragma unroll
    for (int b = 0; b < NUM_BASES; ++b)
      acc = fmaf(c[b], bases[(size_t)b * DIM * DIM + i], acc);
    W[(size_t)r * DIM * DIM + i] = acc;
  }
}

// per-(dst,rel) counts (packed u8) + per-relation histogram
__global__ void count_edges(const int* __restrict__ dst,
                            const int* __restrict__ etype,
                            unsigned int* __restrict__ cnt8,
                            int* __restrict__ hist) {
  int e = blockIdx.x * blockDim.x + threadIdx.x;
  if (e >= NUM_EDGES) return;
  int d = dst[e], r = etype[e];
  unsigned int key = (unsigned)d * NUM_RELATIONS + (unsigned)r;
  atomicAdd(&cnt8[key >> 2], 1u << (8u * (key & 3u)));
  atomicAdd(&hist[r], 1);
}

// serial exclusive scan + build tile -> (relation, local 64-edge tile) table
__global__ void scan_offsets(const int* __restrict__ hist,
                             int* __restrict__ offs,
                             int* __restrict__ curs,
                             int* __restrict__ tile_rel,
                             int* __restrict__ tile_off,
                             int* __restrict__ total_tiles) {
  if (threadIdx.x != 0 || blockIdx.x != 0) return;
  int acc = 0, t = 0;
  for (int r = 0; r < NUM_RELATIONS; ++r) {
    offs[r] = acc;
    curs[r] = acc;
    int h = hist[r];
    int nt = (h + 63) >> 6;
    for (int k = 0; k < nt && t < MAX_TILES; ++k) { tile_rel[t] = r; tile_off[t] = k; ++t; }
    acc += h;
  }
  total_tiles[0] = t;
}

// counting-sort scatter: edge ids grouped by relation
__global__ void scatter_edges(const int* __restrict__ etype,
                              int* __restrict__ curs,
                              int* __restrict__ sorted) {
  int e = blockIdx.x * blockDim.x + threadIdx.x;
  if (e >= NUM_EDGES) return;
  int pos = atomicAdd(&curs[etype[e]], 1);
  sorted[pos] = e;
}

// 64-edge x 128-out WMMA GEMM tile per block; 8 waves each own a 16-col slab,
// each wave carries 4 row-tile accumulators so every B fragment feeds 4 WMMAs.
__global__ __launch_bounds__(256)
void rgcn_msg_wmma(const float* __restrict__ x,
                   const float* __restrict__ W,
                   const int* __restrict__ sorted,
                   const int* __restrict__ srcA,
                   const int* __restrict__ dstA,
                   const int* __restrict__ offs,
                   const int* __restrict__ hist,
                   const unsigned int* __restrict__ cnt8,
                   const int* __restrict__ tile_rel,
                   const int* __restrict__ tile_off,
                   const int* __restrict__ total_tiles,
                   float* __restrict__ agg) {
  int gt = blockIdx.x;
  if (gt >= total_tiles[0]) return;
  int r  = tile_rel[gt];
  int t0 = tile_off[gt] << 6;         // 64 edges per tile
  int segStart = offs[r], segLen = hist[r];

  __shared__ float As[64][132];       // pitch 132 -> conflict-free WMMA column reads
  __shared__ float normS[64];
  __shared__ int   dstS[64];
  __shared__ int   srcS[64];

  int tid = threadIdx.x;
  if (tid < 64) {
    int li = t0 + tid;
    float nrm = 0.f; int d = 0, s = 0;
    if (li < segLen) {
      int e = sorted[segStart + li];
      s = srcA[e]; d = dstA[e];
      unsigned key = (unsigned)d * NUM_RELATIONS + (unsigned)r;
      unsigned w = cnt8[key >> 2];
      unsigned c = (w >> (8u * (key & 3u))) & 0xFFu;
      nrm = 1.0f / fmaxf((float)c, 1.0f);
    }
    normS[tid] = nrm; dstS[tid] = d; srcS[tid] = s;
  }
  __syncthreads();

  { // gather A tile: 64 rows x 128 cols; each thread moves 32 floats (8x float4)
    int row  = tid >> 2;
    int col0 = (tid & 3) * 32;
    const float4* p = reinterpret_cast<const float4*>(&x[(size_t)srcS[row] * DIM + col0]);
#pragma unroll
    for (int j = 0; j < 8; ++j) {
      float4 v = p[j];
      *reinterpret_cast<float4*>(&As[row][col0 + j * 4]) = v;
    }
  }
  __syncthreads();

  int lane  = tid & 31;
  int wv    = tid >> 5;
  int o0    = wv << 4;                // this wave's output-column tile
  int mrow  = lane & 15;              // A: M index / B: N index
  int khalf = (lane >> 4) << 1;       // 0 or 2
  const float* Wr = W + (size_t)r * DIM * DIM;

  v8f acc[4];
  v8f zero = {0.f, 0.f, 0.f, 0.f, 0.f, 0.f, 0.f, 0.f};
#pragma unroll
  for (int rt = 0; rt < 4; ++rt) acc[rt] = zero;

#pragma unroll 4
  for (int k = 0; k < DIM; k += 4) {
    v2f b;
    b[0] = Wr[(size_t)(k + khalf) * DIM + o0 + mrow];
    b[1] = Wr[(size_t)(k + khalf + 1) * DIM + o0 + mrow];
#pragma unroll
    for (int rt = 0; rt < 4; ++rt) {
      v2f a;
      a[0] = As[(rt << 4) + mrow][k + khalf];
      a[1] = As[(rt << 4) + mrow][k + khalf + 1];
      acc[rt] = __builtin_amdgcn_wmma_f32_16x16x4_f32(false, a, false, b, (short)0,
                                                      acc[rt], false, false);
    }
  }

#pragma unroll
  for (int rt = 0; rt < 4; ++rt) {
#pragma unroll
    for (int v = 0; v < 8; ++v) {
      int row = (rt << 4) + v + ((lane >> 4) << 3);   // C layout: M = v + 8*(lane/16)
      float val = acc[rt][v] * normS[row];            // padded rows: norm 0 -> +0.0
      atomicAdd(&agg[(size_t)dstS[row] * DIM + o0 + mrow], val);
    }
  }
}

__global__ void leaky_relu_k(float* __restrict__ a, int n) {
  int i = blockIdx.x * blockDim.x + threadIdx.x;
  if (i < n) { float v = a[i]; a[i] = v > 0.f ? v : 0.01f * v; }
}

// score = sigmoid(h . W_r . t), one 128-thread block per triple
__global__ __launch_bounds__(128)
void score_k(const float* __restrict__ upd,
             const float* __restrict__ W,
             const int* __restrict__ triples,
             float* __restrict__ out) {
  int t = blockIdx.x;
  int h = triples[t * 3 + 0], r = triples[t * 3 + 1], tl = triples[t * 3 + 2];
  __shared__ float hs[DIM];
  __shared__ float red[DIM];
  int tid = threadIdx.x;
  hs[tid] = upd[(size_t)h * DIM + tid];
  __syncthreads();
  const float* Wr = W + (size_t)r * DIM * DIM;
  float acc = 0.f;
#pragma unroll 8
  for (int d = 0; d < DIM; ++d) acc = fmaf(hs[d], Wr[(size_t)d * DIM + tid], acc);
  acc *= upd[(size_t)tl * DIM + tid];
  red[tid] = acc; __syncthreads();
  for (int s = 64; s > 0; s >>= 1) {
    if (tid < s) red[tid] += red[tid + s];
    __syncthreads();
  }
  if (tid == 0) out[t] = 1.0f / (1.0f + expf(-red[0]));
}

extern "C" void kernel_launch(void* const* d_in, const int* in_sizes, int n_in,
                              void* d_out, int out_size, void* d_ws, size_t ws_size,
                              hipStream_t stream) {
  const int*   edge_index = (const int*)d_in[0];     // [2, 300000]
  const int*   edge_type  = (const int*)d_in[1];     // [300000]
  const int*   triples    = (const int*)d_in[2];     // [4096, 3]
  const float* x          = (const float*)d_in[3];   // [100000, 128]
  const float* bases      = (const float*)d_in[4];   // [8, 128, 128]
  const float* coeff      = (const float*)d_in[5];   // [200, 8]
  float* out = (float*)d_out;
  char*  ws  = (char*)d_ws;

  const int* src = edge_index;
  const int* dst = edge_index + NUM_EDGES;

  float*        W      = (float*)(ws + OFF_W);
  unsigned int* cnt8   = (unsigned int*)(ws + OFF_CNT);
  float*        agg    = (float*)(ws + OFF_AGG);
  char*         meta   = ws + OFF_META;
  int* hist   = (int*)(meta + META_HIST);
  int* offs   = (int*)(meta + META_OFFS);
  int* curs   = (int*)(meta + META_CURS);
  int* tott   = (int*)(meta + META_TOTT);
  int* trel   = (int*)(meta + META_TREL);
  int* toff   = (int*)(meta + META_TOFF);
  int* sorted = (int*)(meta + META_SORT);

  // fresh state every call (deterministic, graph-capturable)
  hipMemsetAsync(cnt8, 0, (size_t)20000000, stream);
  hipMemsetAsync(agg, 0, (size_t)NUM_ENTITIES * DIM * sizeof(float), stream);
  hipMemsetAsync(hist, 0, 1024, stream);

  build_weights<<<NUM_RELATIONS, 256, 0, stream>>>(bases, coeff, W);
  count_edges<<<(NUM_EDGES + 255) / 256, 256, 0, stream>>>(dst, edge_type, cnt8, hist);
  scan_offsets<<<1, 32, 0, stream>>>(hist, offs, curs, trel, toff, tott);
  scatter_edges<<<(NUM_EDGES + 255) / 256, 256, 0, stream>>>(edge_type, curs, sorted);
  rgcn_msg_wmma<<<MAX_TILE_BLOCKS, 256, 0, stream>>>(x, W, sorted, src, dst, offs, hist,
                                                     cnt8, trel, toff, tott, agg);
  leaky_relu_k<<<(NUM_ENTITIES * DIM + 255) / 256, 256, 0, stream>>>(agg, NUM_ENTITIES * DIM);
  score_k<<<NUM_TRIPLES, 128, 0, stream>>>(agg, W, triples, out);
}